// FastNGramLM_26706106646904
// MI455X (gfx1250) — compile-verified
//
#include <hip/hip_runtime.h>
#include <stdint.h>

// Batched n-gram LM advance on gfx1250 (MI455X).
// One block per batch row. All 1024 labels of a row share the same backoff
// chain (<=4 states). We async-stage (CDNA5 global->LDS async path):
//   - each small chain state's K~10 arcs  (b32 async copies)
//   - the chain-terminal "big" state's full arc list, <=1024 arcs / 12KB
//     (b128 async copies, one 16B chunk per thread)
// then every label resolves entirely from LDS. Output stored as float4
// (b128) coalesced.

#define LM_VOCAB     1024
#define LM_MAX_ORDER 4
#define ARC_CAP      16     // >= K (=10); power of two
#define BIG_CAP      1024   // max arcs of the big (start) state staged in LDS
#define BLOCK_T      256

__global__ __launch_bounds__(BLOCK_T)
void ngram_advance_kernel(
    const float* __restrict__ arcs_weights,
    const float* __restrict__ backoff_weights,
    const int*   __restrict__ to_states,
    const int*   __restrict__ ilabels,
    const int*   __restrict__ backoff_to_states,
    const int*   __restrict__ state_start_arcs,
    const int*   __restrict__ state_end_arcs,
    const int*   __restrict__ states,
    float* __restrict__ out_scores,   // [B, V]
    float* __restrict__ out_next)     // [B, V] (int values stored as float)
{
    __shared__ int   ch_state[LM_MAX_ORDER];
    __shared__ float ch_cumw[LM_MAX_ORDER];
    __shared__ int   ch_start[LM_MAX_ORDER];
    __shared__ int   ch_len[LM_MAX_ORDER];
    __shared__ int   sh_depth;
    __shared__ int   sh_bigd;                 // depth index staged in big LDS, -1 none
    __shared__ float sh_miss;
    __shared__ int   s_ilab[LM_MAX_ORDER * ARC_CAP];
    __shared__ float s_w   [LM_MAX_ORDER * ARC_CAP];
    __shared__ int   s_to  [LM_MAX_ORDER * ARC_CAP];
    __shared__ __align__(16) int   sb_ilab[BIG_CAP];
    __shared__ __align__(16) float sb_w   [BIG_CAP];
    __shared__ __align__(16) int   sb_to  [BIG_CAP];

    const int b   = blockIdx.x;
    const int tid = threadIdx.x;

    // ---- 1. one thread walks the backoff chain (wave-uniform -> SMEM) ----
    if (tid == 0) {
        int   s  = states[b];
        float cw = 0.0f;
        int   d  = 0;
        int   bigd = -1;
        for (;;) {
            ch_state[d] = s;
            ch_cumw[d]  = cw;
            int st      = state_start_arcs[s];
            int len     = state_end_arcs[s] - st;
            ch_start[d] = st;
            ch_len[d]   = len;
            if (len > ARC_CAP && len <= BIG_CAP && bigd < 0) bigd = d;
            ++d;
            if (s == 0 || d == LM_MAX_ORDER) break;
            cw += backoff_weights[s];
            s   = backoff_to_states[s];
        }
        sh_depth = d;
        sh_bigd  = bigd;
        // full-miss score (state 0 backoff weight is 0 -> closed form matches
        // the MAX_ORDER-iteration reference loop exactly).
        sh_miss  = ch_cumw[d - 1] + backoff_weights[ch_state[d - 1]];
    }
    __syncthreads();

    const int depth = sh_depth;
    const int bigd  = sh_bigd;

    // ---- 2a. async-stage small arc lists (gfx1250 async b32 path) ----
    if (tid < depth * ARC_CAP) {
        const int d   = tid >> 4;              // ARC_CAP == 16
        const int j   = tid & (ARC_CAP - 1);
        const int len = ch_len[d];
        if (len <= ARC_CAP && j < len) {
            const int ai = ch_start[d] + j;
            uint32_t l0 = (uint32_t)(uintptr_t)&s_ilab[tid];
            uint32_t l1 = (uint32_t)(uintptr_t)&s_w[tid];
            uint32_t l2 = (uint32_t)(uintptr_t)&s_to[tid];
            uint64_t g0 = (uint64_t)(uintptr_t)(ilabels      + ai);
            uint64_t g1 = (uint64_t)(uintptr_t)(arcs_weights + ai);
            uint64_t g2 = (uint64_t)(uintptr_t)(to_states    + ai);
            asm volatile("global_load_async_to_lds_b32 %0, %1, off"
                         :: "v"(l0), "v"(g0) : "memory");
            asm volatile("global_load_async_to_lds_b32 %0, %1, off"
                         :: "v"(l1), "v"(g1) : "memory");
            asm volatile("global_load_async_to_lds_b32 %0, %1, off"
                         :: "v"(l2), "v"(g2) : "memory");
        }
    }

    // ---- 2b. async-stage the big terminal state (b128 wide async path) ----
    if (bigd >= 0) {
        const int st  = ch_start[bigd];
        const int len = ch_len[bigd];
        if ((st & 3) == 0) {
            const int nch = len >> 2;          // 16B chunks
            for (int c = tid; c < nch; c += BLOCK_T) {
                const int e = c << 2;
                uint32_t l0 = (uint32_t)(uintptr_t)&sb_ilab[e];
                uint32_t l1 = (uint32_t)(uintptr_t)&sb_w[e];
                uint32_t l2 = (uint32_t)(uintptr_t)&sb_to[e];
                uint64_t g0 = (uint64_t)(uintptr_t)(ilabels      + st + e);
                uint64_t g1 = (uint64_t)(uintptr_t)(arcs_weights + st + e);
                uint64_t g2 = (uint64_t)(uintptr_t)(to_states    + st + e);
                asm volatile("global_load_async_to_lds_b128 %0, %1, off"
                             :: "v"(l0), "v"(g0) : "memory");
                asm volatile("global_load_async_to_lds_b128 %0, %1, off"
                             :: "v"(l1), "v"(g1) : "memory");
                asm volatile("global_load_async_to_lds_b128 %0, %1, off"
                             :: "v"(l2), "v"(g2) : "memory");
            }
            for (int j = (nch << 2) + tid; j < len; j += BLOCK_T) {
                uint32_t l0 = (uint32_t)(uintptr_t)&sb_ilab[j];
                uint32_t l1 = (uint32_t)(uintptr_t)&sb_w[j];
                uint32_t l2 = (uint32_t)(uintptr_t)&sb_to[j];
                uint64_t g0 = (uint64_t)(uintptr_t)(ilabels      + st + j);
                uint64_t g1 = (uint64_t)(uintptr_t)(arcs_weights + st + j);
                uint64_t g2 = (uint64_t)(uintptr_t)(to_states    + st + j);
                asm volatile("global_load_async_to_lds_b32 %0, %1, off"
                             :: "v"(l0), "v"(g0) : "memory");
                asm volatile("global_load_async_to_lds_b32 %0, %1, off"
                             :: "v"(l1), "v"(g1) : "memory");
                asm volatile("global_load_async_to_lds_b32 %0, %1, off"
                             :: "v"(l2), "v"(g2) : "memory");
            }
        } else {                               // unaligned generic fallback
            for (int j = tid; j < len; j += BLOCK_T) {
                uint32_t l0 = (uint32_t)(uintptr_t)&sb_ilab[j];
                uint32_t l1 = (uint32_t)(uintptr_t)&sb_w[j];
                uint32_t l2 = (uint32_t)(uintptr_t)&sb_to[j];
                uint64_t g0 = (uint64_t)(uintptr_t)(ilabels      + st + j);
                uint64_t g1 = (uint64_t)(uintptr_t)(arcs_weights + st + j);
                uint64_t g2 = (uint64_t)(uintptr_t)(to_states    + st + j);
                asm volatile("global_load_async_to_lds_b32 %0, %1, off"
                             :: "v"(l0), "v"(g0) : "memory");
                asm volatile("global_load_async_to_lds_b32 %0, %1, off"
                             :: "v"(l1), "v"(g1) : "memory");
                asm volatile("global_load_async_to_lds_b32 %0, %1, off"
                             :: "v"(l2), "v"(g2) : "memory");
            }
        }
    }
    asm volatile("s_wait_asynccnt 0" ::: "memory");
    __syncthreads();

    // ---- 3. resolve 4 consecutive labels per thread, store as b128 ----
    for (int v0 = tid << 2; v0 < LM_VOCAB; v0 += BLOCK_T << 2) {
        float4 sc, nx;
        #pragma unroll
        for (int k = 0; k < 4; ++k) {
            const int v = v0 + k;
            float score = sh_miss;
            float nxt   = 0.0f;
            for (int d = 0; d < depth; ++d) {
                const int len = ch_len[d];
                int   to  = 0;
                float w   = 0.0f;
                bool  hit = false;
                if (len <= ARC_CAP) {
                    const int base = d << 4;
                    for (int j = 0; j < len; ++j) {
                        if (s_ilab[base + j] == v) {
                            hit = true; w = s_w[base + j]; to = s_to[base + j];
                            break;
                        }
                    }
                } else if (d == bigd) {
                    // dense direct probe first (start state is exactly dense)
                    const int off = v - sb_ilab[0];
                    if (off >= 0 && off < len && sb_ilab[off] == v) {
                        hit = true; w = sb_w[off]; to = sb_to[off];
                    } else {
                        int lo = 0, hi = len;
                        while (lo < hi) {
                            const int mid = (lo + hi) >> 1;
                            if (sb_ilab[mid] < v) lo = mid + 1; else hi = mid;
                        }
                        if (lo < len && sb_ilab[lo] == v) {
                            hit = true; w = sb_w[lo]; to = sb_to[lo];
                        }
                    }
                } else {
                    // oversized state not staged: global binary search
                    const int st = ch_start[d];
                    int lo = st, hi = st + len;
                    while (lo < hi) {
                        const int mid = (lo + hi) >> 1;
                        if (ilabels[mid] < v) lo = mid + 1; else hi = mid;
                    }
                    if (lo < st + len && ilabels[lo] == v) {
                        hit = true; w = arcs_weights[lo]; to = to_states[lo];
                    }
                }
                if (hit) {
                    score = ch_cumw[d] + w;
                    nxt   = (float)to;
                    break;
                }
            }
            ((float*)&sc)[k] = score;
            ((float*)&nx)[k] = nxt;
        }
        const size_t idx = (size_t)b * LM_VOCAB + (size_t)v0;
        *(float4*)(out_scores + idx) = sc;
        *(float4*)(out_next   + idx) = nx;
    }
}

extern "C" void kernel_launch(void* const* d_in, const int* in_sizes, int n_in,
                              void* d_out, int out_size, void* d_ws, size_t ws_size,
                              hipStream_t stream) {
    // setup_inputs() order:
    //  0 arcs_weights f32 | 1 backoff_weights f32 | 2 from_states i32 (unused)
    //  3 to_states i32    | 4 ilabels i32         | 5 backoff_to_states i32
    //  6 state_start_arcs i32 | 7 state_end_arcs i32 | 8 state_order i32 (unused)
    //  9 states i32 [B]
    const float* arcs_weights      = (const float*)d_in[0];
    const float* backoff_weights   = (const float*)d_in[1];
    const int*   to_states         = (const int*)d_in[3];
    const int*   ilabels           = (const int*)d_in[4];
    const int*   backoff_to_states = (const int*)d_in[5];
    const int*   state_start_arcs  = (const int*)d_in[6];
    const int*   state_end_arcs    = (const int*)d_in[7];
    const int*   states            = (const int*)d_in[9];

    const int B = in_sizes[9];
    float* out_scores = (float*)d_out;                        // first B*V floats
    float* out_next   = (float*)d_out + (size_t)B * LM_VOCAB; // second B*V

    ngram_advance_kernel<<<B, BLOCK_T, 0, stream>>>(
        arcs_weights, backoff_weights, to_states, ilabels,
        backoff_to_states, state_start_arcs, state_end_arcs, states,
        out_scores, out_next);
}